// S2Layer_60833916780946
// MI455X (gfx1250) — compile-verified
//
#include <hip/hip_runtime.h>
#include <math.h>

// ---------------- static config (mirrors reference) ----------------
#define B_IN   32
#define B_OUT  16
#define F_IN   6
#define F_OUT  32
#define BATCH  32
#define NGRID  24
#define NSPEC  256          // B_OUT^2
#define NB_IN  64           // 2*B_IN
#define NB_OUT 32           // 2*B_OUT
#define NPAIR  5456         // sum_{l<16} (2l+1)^2
#define PI_F   3.14159265358979323846f

typedef float v2f __attribute__((ext_vector_type(2)));
typedef float v8f __attribute__((ext_vector_type(8)));

// pairs below degree l:  sum_{l'<l} (2l'+1)^2 = l(4l^2-1)/3
__device__ __host__ __forceinline__ int pb_l(int l) { return l * (4 * l * l - 1) / 3; }

// Wigner small-d  d^l_{mp,mm}(beta), direct lgamma formula (matches reference)
__device__ float wig_d(int l, int mp, int mm, float beta) {
  float cb = cosf(0.5f * beta), sb = sinf(0.5f * beta);
  float lognorm = 0.5f * (lgammaf((float)(l + mm + 1)) + lgammaf((float)(l - mm + 1)) +
                          lgammaf((float)(l + mp + 1)) + lgammaf((float)(l - mp + 1)));
  int s0 = max(0, mm - mp), s1 = min(l + mm, l - mp);
  float acc = 0.f;
  for (int s = s0; s <= s1; ++s) {
    float logden = lgammaf((float)(l + mm - s + 1)) + lgammaf((float)(s + 1)) +
                   lgammaf((float)(mp - mm + s + 1)) + lgammaf((float)(l - mp - s + 1));
    float sgn = ((mp - mm + s) & 1) ? -1.f : 1.f;
    acc += sgn * expf(lognorm - logden)
              * powf(cb, (float)(2 * l + mm - mp - 2 * s))
              * powf(sb, (float)(mp - mm + 2 * s));
  }
  return acc;
}

__device__ __forceinline__ void spec_decode(int spec, int& l, int& mp) {
  l = (int)sqrtf((float)spec);
  while ((l + 1) * (l + 1) <= spec) ++l;
  while (l * l > spec) --l;
  mp = spec - l * l - l;
}

// ---------------- constant-synthesis kernels ----------------
__global__ void k_qw(float* qw) {
  int j = threadIdx.x;
  if (j >= NB_IN) return;
  float s = 0.f;
  for (int k = 0; k < B_IN; ++k)
    s += sinf((float)((2 * j + 1) * (2 * k + 1)) * PI_F / (4.f * B_IN)) / (float)(2 * k + 1);
  float th = PI_F * (2 * j + 1) / (4.f * B_IN);
  qw[j] = (2.f / B_IN) * sinf(th) * s * (2.f * PI_F / (2.f * B_IN));
}

// 32x32 IDFT tables + 16x64 alpha-DFT tables
__global__ void k_dfttab(float* ctab, float* stab, float* d64c, float* d64s) {
  int idx = blockIdx.x * blockDim.x + threadIdx.x;
  if (idx < 1024) {
    int a = idx >> 5, m = idx & 31;
    float ang = 2.f * PI_F * (float)(a * m) / 32.f;
    ctab[idx] = cosf(ang); stab[idx] = sinf(ang);
  } else if (idx < 2048) {
    int t = idx - 1024;
    int m = t >> 6, a = t & 63;
    float ang = 2.f * PI_F * (float)(m * a) / 64.f;
    d64c[t] = cosf(ang); d64s[t] = sinf(ang);
  }
}

// ws2[spec][b] = qw[b] * d^l_{mp,0}(beta_in[b])
__global__ void k_ws2(const float* qw, float* ws2) {
  int idx = blockIdx.x * blockDim.x + threadIdx.x;
  if (idx >= NSPEC * NB_IN) return;
  int spec = idx >> 6, b = idx & 63;
  int l, mp; spec_decode(spec, l, mp);
  float beta = PI_F * (2 * b + 1) / (4.f * B_IN);
  ws2[idx] = qw[b] * wig_d(l, mp, 0, beta);
}

// wso3 block l: [b][i][j] = (2l+1) * d^l_{i-l,j-l}(beta_out[b]) at 32*pb(l)
__global__ void k_wso3(float* wso3) {
  int idx = blockIdx.x * blockDim.x + threadIdx.x;
  if (idx >= NPAIR * NB_OUT) return;
  int b = idx & 31, pair = idx >> 5;
  int l = 0; while (l < 15 && pb_l(l + 1) <= pair) ++l;
  int L = 2 * l + 1, r = pair - pb_l(l);
  int i = r / L, j = r % L;
  float beta = PI_F * (2 * b + 1) / (4.f * B_OUT);
  wso3[32 * pb_l(l) + (b * L + i) * L + j] = (float)L * wig_d(l, i - l, j - l, beta);
}

// fk[g][spec] = exp(-i mp alpha_g) * d^l_{mp,0}(beta_g)
__global__ void k_fk(float* fkre, float* fkim) {
  int idx = blockIdx.x * blockDim.x + threadIdx.x;
  if (idx >= NGRID * NSPEC) return;
  int g = idx >> 8, spec = idx & 255;
  int bi = g >> 3, ai = g & 7;
  float beta = (float)(bi + 1) * (PI_F / 8.f) / 3.f;
  float alpha = (float)ai * (2.f * PI_F / 8.f);
  int l, mp; spec_decode(spec, l, mp);
  float d0 = wig_d(l, mp, 0, beta);
  float ang = (float)mp * alpha;
  fkre[idx] = cosf(ang) * d0;
  fkim[idx] = -sinf(ang) * d0;
}

// conj(psi)[spec][f][o] = SCALING * sum_g kernel[f][o][g] * conj(fk[g][spec])
__global__ void k_cpsi(const float* __restrict__ ker, const float* __restrict__ fkre,
                       const float* __restrict__ fkim, float* cpre, float* cpim, float scaling) {
  int idx = blockIdx.x * blockDim.x + threadIdx.x;
  if (idx >= NSPEC * F_IN * F_OUT) return;
  int o = idx & 31, t = idx >> 5;
  int f = t % F_IN, spec = t / F_IN;
  float re = 0.f, im = 0.f;
  for (int g = 0; g < NGRID; ++g) {
    float kv = ker[(f * F_OUT + o) * NGRID + g];
    re += kv * fkre[g * NSPEC + spec];
    im -= kv * fkim[g * NSPEC + spec];   // conjugate
  }
  cpre[idx] = re * scaling;
  cpim[idx] = im * scaling;
}

// ---------------- front end ----------------
// alpha-DFT of real x for m=0..15: xf[z][f][b][m]
__global__ void k_xf(const float* __restrict__ x, const float* __restrict__ d64c,
                     const float* __restrict__ d64s, float* xfre, float* xfim) {
  int idx = blockIdx.x * blockDim.x + threadIdx.x;
  if (idx >= BATCH * F_IN * NB_IN * 16) return;
  int m = idx & 15, t = idx >> 4;
  int b = t & 63; t >>= 6;
  int f = t % F_IN, z = t / F_IN;
  const float* xr = x + ((size_t)(z * F_IN + f) * NB_IN + b) * NB_IN;
  float re = 0.f, im = 0.f;
  for (int a = 0; a < NB_IN; ++a) {
    float xv = xr[a];
    re += xv * d64c[m * 64 + a];
    im -= xv * d64s[m * 64 + a];
  }
  xfre[idx] = re; xfim[idx] = im;
}

// xhat[spec][z][f] = sum_b ws2[spec][b] * xf[z][f][b][mp]   (conj for mp<0)
__global__ void k_xhat(const float* __restrict__ ws2, const float* __restrict__ xfre,
                       const float* __restrict__ xfim, float* xhre, float* xhim) {
  int idx = blockIdx.x * blockDim.x + threadIdx.x;
  if (idx >= NSPEC * BATCH * F_IN) return;
  int f = idx % F_IN, t = idx / F_IN;
  int z = t & 31, spec = t >> 5;
  int l, mp; spec_decode(spec, l, mp);
  int am = mp < 0 ? -mp : mp;
  float sg = mp < 0 ? -1.f : 1.f;
  float re = 0.f, im = 0.f;
  for (int b = 0; b < NB_IN; ++b) {
    float w = ws2[spec * NB_IN + b];
    int q = ((z * F_IN + f) * NB_IN + b) * 16 + am;
    re += w * xfre[q];
    im += w * sg * xfim[q];
  }
  xhre[idx] = re; xhim[idx] = im;
}

// zl[pair][o] = sum_f xhat[l^2+i][z][f] * cpsi[l^2+j][f][o]   (complex)
__global__ void k_zl(const float* __restrict__ xhre, const float* __restrict__ xhim,
                     const float* __restrict__ cpre, const float* __restrict__ cpim,
                     float* zlre, float* zlim, int z) {
  int idx = blockIdx.x * blockDim.x + threadIdx.x;
  if (idx >= NPAIR * F_OUT) return;
  int o = idx & 31, pair = idx >> 5;
  int l = 0; while (l < 15 && pb_l(l + 1) <= pair) ++l;
  int L = 2 * l + 1, r = pair - pb_l(l);
  int i = r / L, j = r % L;
  int specm = l * l + i, specn = l * l + j;
  float re = 0.f, im = 0.f;
  for (int f = 0; f < F_IN; ++f) {
    float ar = xhre[(specm * BATCH + z) * F_IN + f];
    float ai = xhim[(specm * BATCH + z) * F_IN + f];
    float cr = cpre[(specn * F_IN + f) * F_OUT + o];
    float ci = cpim[(specn * F_IN + f) * F_OUT + o];
    re += ar * cr - ai * ci;
    im += ar * ci + ai * cr;
  }
  zlre[idx] = re; zlim[idx] = im;
}

// ---------------- fused per-plane back end ----------------
// One block = one (o,b) plane of one batch element:
//   Phase 1: synthesize fhat(m,n) into LDS (gather over degrees l; zl/wso3 are L2-resident)
//   Phase 2: WMMA stage 1  T1 = F * E  (over n -> gamma), complex, T1 kept in LDS (buffer reuse)
//   Phase 3: WMMA stage 2  f = Re(E^T * T1) (over m -> alpha) + bias -> global output
// fhat and T1 never touch HBM: saves ~1 GB of round trips per call vs the unfused pipeline.
__global__ void __launch_bounds__(128)
k_plane(const float* __restrict__ wso3, const float* __restrict__ zlre,
        const float* __restrict__ zlim, const float* __restrict__ ctab,
        const float* __restrict__ stab, const float* __restrict__ bias,
        float* __restrict__ out, int z) {
  __shared__ float sRe[1024];
  __shared__ float sIm[1024];
  int p = blockIdx.x;                 // 1024 planes = (o,b)
  int o = p >> 5, b = p & 31;
  int tid = threadIdx.x;

  // ---- Phase 1: fhat plane into LDS ----
  for (int e = tid; e < 1024; e += 128) {
    int m = e >> 5, n = e & 31;
    int ms = (m < 16) ? m : m - 32;
    int ns = (n < 16) ? n : n - 32;
    int am = ms < 0 ? -ms : ms, an = ns < 0 ? -ns : ns;
    int lmin = am > an ? am : an;
    float re = 0.f, im = 0.f;
    for (int l = lmin; l < 16; ++l) {
      int L = 2 * l + 1, i = ms + l, j = ns + l;
      float w = wso3[32 * pb_l(l) + (b * L + i) * L + j];
      int q = (pb_l(l) + i * L + j) * F_OUT + o;
      re += w * zlre[q];
      im += w * zlim[q];
    }
    sRe[e] = re; sIm[e] = im;
  }
  __syncthreads();

  int wave = tid >> 5, lane = tid & 31;
  int mt = (wave >> 1) << 4, gt = (wave & 1) << 4;   // 16x16 tile of this wave
  int half = lane >> 4, lr = lane & 15;

  // ---- Phase 2: stage-1 WMMA (complex F x complex E) ----
  // A frag layout: lanes<16 hold K=k0,k0+1 of row M=mt+lr; lanes>=16 hold K=k0+2,k0+3
  const float* Ar = sRe + (mt + lr) * 32;
  const float* Ai = sIm + (mt + lr) * 32;
  v8f accR = {0, 0, 0, 0, 0, 0, 0, 0};
  v8f accI = {0, 0, 0, 0, 0, 0, 0, 0};
#pragma unroll
  for (int k0 = 0; k0 < 32; k0 += 4) {
    int ka = k0 + (half << 1);
    v2f ar, ai, bc, bs, bns;
    ar.x = Ar[ka];  ar.y = Ar[ka + 1];
    ai.x = Ai[ka];  ai.y = Ai[ka + 1];
    bc.x = ctab[ka * 32 + gt + lr];  bc.y = ctab[(ka + 1) * 32 + gt + lr];
    bs.x = stab[ka * 32 + gt + lr];  bs.y = stab[(ka + 1) * 32 + gt + lr];
    bns.x = -bs.x; bns.y = -bs.y;    // f32 WMMA has no A/B negate modifier
    accR = __builtin_amdgcn_wmma_f32_16x16x4_f32(false, ar, false, bc,  (short)0, accR, false, false);
    accR = __builtin_amdgcn_wmma_f32_16x16x4_f32(false, ai, false, bns, (short)0, accR, false, false);
    accI = __builtin_amdgcn_wmma_f32_16x16x4_f32(false, ar, false, bs,  (short)0, accI, false, false);
    accI = __builtin_amdgcn_wmma_f32_16x16x4_f32(false, ai, false, bc,  (short)0, accI, false, false);
  }
  __syncthreads();                    // all reads of F done -> reuse LDS for T1
#pragma unroll
  for (int r = 0; r < 8; ++r) {       // D: VGPR r -> M = r + 8*half
    int row = mt + r + (half << 3), col = gt + lr;
    sRe[row * 32 + col] = accR[r];
    sIm[row * 32 + col] = accI[r];
  }
  __syncthreads();

  // ---- Phase 3: stage-2 WMMA (real part of E^T x T1) + bias -> out ----
  const float* Crow = ctab + (mt + lr) * 32;   // A = C[a][m], symmetric table; at == mt
  const float* Srow = stab + (mt + lr) * 32;
  v8f acc = {0, 0, 0, 0, 0, 0, 0, 0};
#pragma unroll
  for (int k0 = 0; k0 < 32; k0 += 4) {
    int ka = k0 + (half << 1);
    v2f ac, ans, br, bi;
    ac.x  =  Crow[ka];  ac.y  =  Crow[ka + 1];
    ans.x = -Srow[ka];  ans.y = -Srow[ka + 1];
    br.x = sRe[ka * 32 + gt + lr];  br.y = sRe[(ka + 1) * 32 + gt + lr];
    bi.x = sIm[ka * 32 + gt + lr];  bi.y = sIm[(ka + 1) * 32 + gt + lr];
    acc = __builtin_amdgcn_wmma_f32_16x16x4_f32(false, ac,  false, br, (short)0, acc, false, false);
    acc = __builtin_amdgcn_wmma_f32_16x16x4_f32(false, ans, false, bi, (short)0, acc, false, false);
  }
  float bv = bias[o];
  size_t base = (((size_t)z * F_OUT + o) * NB_OUT + b) * 1024;
#pragma unroll
  for (int r = 0; r < 8; ++r) {
    int row = mt + r + (half << 3), col = gt + lr;
    out[base + row * 32 + col] = acc[r] + bv;
  }
}

// ---------------- BatchNorm (training stats) + ReLU ----------------
__global__ void k_stats(const float* __restrict__ y, float* psum, float* psqs) {
  int bid = blockIdx.x;
  int o = bid >> 5, zz = bid & 31;
  size_t base = ((size_t)zz * F_OUT + o) * 32768;
  __shared__ float ss[256], sq[256];
  float s = 0.f, s2 = 0.f;
  for (int i = threadIdx.x; i < 32768; i += 256) {
    float v = y[base + i];
    s += v; s2 += v * v;
  }
  ss[threadIdx.x] = s; sq[threadIdx.x] = s2;
  for (int st = 128; st > 0; st >>= 1) {
    __syncthreads();
    if (threadIdx.x < st) { ss[threadIdx.x] += ss[threadIdx.x + st]; sq[threadIdx.x] += sq[threadIdx.x + st]; }
  }
  __syncthreads();
  if (threadIdx.x == 0) { psum[o * 32 + zz] = ss[0]; psqs[o * 32 + zz] = sq[0]; }
}

__global__ void k_final(const float* psum, const float* psqs, const float* gamma,
                        const float* beta, float* scale, float* shift) {
  int o = threadIdx.x;
  if (o >= F_OUT) return;
  float s = 0.f, s2 = 0.f;
  for (int zz = 0; zz < 32; ++zz) { s += psum[o * 32 + zz]; s2 += psqs[o * 32 + zz]; }
  const float n = 1048576.f;
  float mean = s / n;
  float var = fmaxf(s2 / n - mean * mean, 0.f);
  float inv = rsqrtf(var + 1e-5f);
  float sc = gamma[o] * inv;
  scale[o] = sc;
  shift[o] = beta[o] - mean * sc;
}

__global__ void k_norm(float* y, const float* __restrict__ scale, const float* __restrict__ shift) {
  size_t idx = (size_t)blockIdx.x * blockDim.x + threadIdx.x;
  int o = (int)((idx >> 15) & 31);
  float v = y[idx];
  y[idx] = fmaxf(fmaf(v, scale[o], shift[o]), 0.f);
}

// ---------------- host launcher ----------------
extern "C" void kernel_launch(void* const* d_in, const int* in_sizes, int n_in,
                              void* d_out, int out_size, void* d_ws, size_t ws_size,
                              hipStream_t stream) {
  (void)in_sizes; (void)n_in; (void)out_size; (void)ws_size;
  const float* x     = (const float*)d_in[0];
  const float* ker   = (const float*)d_in[1];
  const float* bias  = (const float*)d_in[2];
  const float* gamma = (const float*)d_in[3];
  const float* betab = (const float*)d_in[4];
  float* out = (float*)d_out;

  // workspace carve-up (256B aligned) -- ~5 MB total after fusion
  size_t off = 0;
  auto alloc = [&](size_t nf) { size_t o = off; off += ((nf * 4 + 255) / 256) * 256; return o; };
  char* ws = (char*)d_ws;
#define WSPTR(name, nf) float* name = (float*)(ws + alloc(nf))
  WSPTR(qw,    64);
  WSPTR(ctab,  1024);   WSPTR(stab, 1024);
  WSPTR(d64c,  1024);   WSPTR(d64s, 1024);
  WSPTR(ws2,   NSPEC * NB_IN);
  WSPTR(fkre,  NGRID * NSPEC);          WSPTR(fkim, NGRID * NSPEC);
  WSPTR(cpre,  NSPEC * F_IN * F_OUT);   WSPTR(cpim, NSPEC * F_IN * F_OUT);
  WSPTR(xfre,  BATCH * F_IN * NB_IN * 16); WSPTR(xfim, BATCH * F_IN * NB_IN * 16);
  WSPTR(xhre,  NSPEC * BATCH * F_IN);   WSPTR(xhim, NSPEC * BATCH * F_IN);
  WSPTR(wso3,  NB_OUT * NPAIR);
  WSPTR(zlre,  NPAIR * F_OUT);          WSPTR(zlim, NPAIR * F_OUT);
  WSPTR(psum,  1024);                   WSPTR(psqs, 1024);
  WSPTR(scale, 32);                     WSPTR(shift, 32);
#undef WSPTR

  const float SCALING = 1.0f / 96.0f;   // 1/sqrt(NGRID*F_IN*B_OUT^4/B_IN^2)

  // constants
  k_qw<<<1, 64, 0, stream>>>(qw);
  k_dfttab<<<8, 256, 0, stream>>>(ctab, stab, d64c, d64s);
  k_ws2<<<64, 256, 0, stream>>>(qw, ws2);
  k_wso3<<<682, 256, 0, stream>>>(wso3);
  k_fk<<<24, 256, 0, stream>>>(fkre, fkim);
  k_cpsi<<<192, 256, 0, stream>>>(ker, fkre, fkim, cpre, cpim, SCALING);

  // front end
  k_xf<<<768, 256, 0, stream>>>(x, d64c, d64s, xfre, xfim);
  k_xhat<<<192, 256, 0, stream>>>(ws2, xfre, xfim, xhre, xhim);

  // per-batch-element back end: zl mixing + fused fhat/IDFT2D plane kernel
  for (int z = 0; z < BATCH; ++z) {
    k_zl<<<682, 256, 0, stream>>>(xhre, xhim, cpre, cpim, zlre, zlim, z);
    k_plane<<<1024, 128, 0, stream>>>(wso3, zlre, zlim, ctab, stab, bias, out, z);
  }

  // BatchNorm (training-mode stats) + ReLU, deterministic tree reductions
  k_stats<<<1024, 256, 0, stream>>>(out, psum, psqs);
  k_final<<<1, 32, 0, stream>>>(psum, psqs, gamma, betab, scale, shift);
  k_norm<<<131072, 256, 0, stream>>>(out, scale, shift);
}